// DecoderGRU_82772609728720
// MI455X (gfx1250) — compile-verified
//
#include <hip/hip_runtime.h>
#include <cmath>

// Problem constants (from reference)
constexpr int V_ = 32000;
constexpr int E_ = 256;
constexpr int H_ = 512;
constexpr int B_ = 32;
constexpr int T_ = 64;
constexpr int SOS = 1;

typedef float v2f __attribute__((ext_vector_type(2)));
typedef float v8f __attribute__((ext_vector_type(8)));

// ---------------------------------------------------------------------------
// Kernel 1: full 2-layer GRU recurrence. One block per batch row (recurrence
// is independent per b). 512 threads: thread j owns hidden element j.
// Weights stream from L2 (10.5MB total, fully L2-resident); x/h state in LDS.
// ---------------------------------------------------------------------------
__device__ __forceinline__ float dotrow(const float* __restrict__ w,
                                        const float* xs, int K) {
  float acc = 0.f;
  #pragma unroll 4
  for (int k = 0; k < K; k += 4) {
    float4 wv = *(const float4*)(w + k);
    acc = fmaf(wv.x, xs[k + 0], acc);
    acc = fmaf(wv.y, xs[k + 1], acc);
    acc = fmaf(wv.z, xs[k + 2], acc);
    acc = fmaf(wv.w, xs[k + 3], acc);
  }
  return acc;
}

__global__ __launch_bounds__(512)
void gru_seq_kernel(const float* __restrict__ enc_hidden,   // [L,B,H]
                    const int*   __restrict__ target,       // [B,T]
                    const float* __restrict__ emb,          // [V,E]
                    const float* __restrict__ Wih0, const float* __restrict__ Whh0,
                    const float* __restrict__ bih0, const float* __restrict__ bhh0,
                    const float* __restrict__ Wih1, const float* __restrict__ Whh1,
                    const float* __restrict__ bih1, const float* __restrict__ bhh1,
                    float* __restrict__ tops,               // ws: [T,B,H]
                    float* __restrict__ hfinal)             // [L,B,H]
{
  __shared__ float xs[H_];
  __shared__ float h0s[H_];
  __shared__ float h1s[H_];
  const int b = blockIdx.x;
  const int j = threadIdx.x;

  h0s[j] = enc_hidden[(size_t)b * H_ + j];
  h1s[j] = enc_hidden[(size_t)(B_ + b) * H_ + j];
  __syncthreads();

  for (int t = 0; t < T_; ++t) {
    // teacher forcing: SOS at t=0, else target[b, t-1]; embedding + ReLU
    const int tok = (t == 0) ? SOS : target[b * T_ + t - 1];
    if (j < E_) xs[j] = fmaxf(emb[(size_t)tok * E_ + j], 0.f);
    __syncthreads();

    // ---- layer 0 (x: E dims, h: H dims) ----
    {
      const float ir = bih0[j]          + dotrow(Wih0 + (size_t)j * E_,            xs,  E_);
      const float iz = bih0[H_ + j]     + dotrow(Wih0 + (size_t)(H_ + j) * E_,     xs,  E_);
      const float in_= bih0[2*H_ + j]   + dotrow(Wih0 + (size_t)(2*H_ + j) * E_,   xs,  E_);
      const float hr = bhh0[j]          + dotrow(Whh0 + (size_t)j * H_,            h0s, H_);
      const float hz = bhh0[H_ + j]     + dotrow(Whh0 + (size_t)(H_ + j) * H_,     h0s, H_);
      const float hn = bhh0[2*H_ + j]   + dotrow(Whh0 + (size_t)(2*H_ + j) * H_,   h0s, H_);
      const float r = 1.f / (1.f + __expf(-(ir + hr)));
      const float z = 1.f / (1.f + __expf(-(iz + hz)));
      const float n = tanhf(in_ + r * hn);
      const float h0new = (1.f - z) * n + z * h0s[j];
      __syncthreads();           // all reads of h0s done
      h0s[j] = h0new;
      __syncthreads();           // new h0 visible (layer-1 input)
    }

    // ---- layer 1 (x = h0, both K = H) ----
    {
      const float ir = bih1[j]          + dotrow(Wih1 + (size_t)j * H_,            h0s, H_);
      const float iz = bih1[H_ + j]     + dotrow(Wih1 + (size_t)(H_ + j) * H_,     h0s, H_);
      const float in_= bih1[2*H_ + j]   + dotrow(Wih1 + (size_t)(2*H_ + j) * H_,   h0s, H_);
      const float hr = bhh1[j]          + dotrow(Whh1 + (size_t)j * H_,            h1s, H_);
      const float hz = bhh1[H_ + j]     + dotrow(Whh1 + (size_t)(H_ + j) * H_,     h1s, H_);
      const float hn = bhh1[2*H_ + j]   + dotrow(Whh1 + (size_t)(2*H_ + j) * H_,   h1s, H_);
      const float r = 1.f / (1.f + __expf(-(ir + hr)));
      const float z = 1.f / (1.f + __expf(-(iz + hz)));
      const float n = tanhf(in_ + r * hn);
      const float h1new = (1.f - z) * n + z * h1s[j];
      __syncthreads();
      h1s[j] = h1new;
      tops[((size_t)t * B_ + b) * H_ + j] = h1new;
      __syncthreads();
    }
  }
  hfinal[(size_t)b * H_ + j]        = h0s[j];
  hfinal[(size_t)(B_ + b) * H_ + j] = h1s[j];
}

// ---------------------------------------------------------------------------
// Kernel 2: logits = tops @ W_out^T + b_out, fused with log_softmax over V.
// fp32 WMMA (V_WMMA_F32_16X16X4_F32). One block per 16-row M-tile; 16 waves
// sweep the 2000 N-tiles and store biased logits; then wave-per-row stats
// (the 2MB tile is L2-resident) and a coalesced normalize pass.
// A tile staged to LDS via global_load_async_to_lds_b128 (ASYNCcnt path).
// ---------------------------------------------------------------------------
__global__ __launch_bounds__(512)
void proj_logsoftmax_kernel(const float* __restrict__ tops,  // [T,B,H]
                            const float* __restrict__ Wout,  // [V,H]
                            const float* __restrict__ bout,  // [V]
                            float* __restrict__ out)         // [B,T,V]
{
  __shared__ float ldsA[16][H_ + 2];     // +2 pad: kill 16-way bank conflicts
  __shared__ float rowlogz[16];

  const int tid  = threadIdx.x;
  const int wave = tid >> 5;
  const int lane = tid & 31;
  const int half = lane >> 4;            // WMMA lane-half (K/M split)
  const int r    = lane & 15;
  const int m0   = blockIdx.x * 16;      // GEMM row m = b*T + t  ([B,T,V] order)

  // --- Stage A tile (16 x 512 f32) via async global->LDS (B128 chunks). ---
  // 8192 floats = 2048 x b128; 512 threads x 4 iterations. Padding is per-row
  // so each 4-float chunk stays contiguous on both sides.
  for (int idx = tid; idx < (16 * H_) / 4; idx += 512) {
    const int e  = idx * 4;
    const int i  = e >> 9;               // row (H_ == 512)
    const int h  = e & (H_ - 1);
    const int m  = m0 + i;
    const int bb = m / T_;
    const int tt = m % T_;
    const float* src = tops + ((size_t)tt * B_ + bb) * H_ + h;
    const unsigned lds_off = (unsigned)(uintptr_t)(&ldsA[i][h]);
    asm volatile("global_load_async_to_lds_b128 %0, %1, off"
                 :: "v"(lds_off), "v"(src) : "memory");
  }
  asm volatile("s_wait_asynccnt 0x0" ::: "memory");
  __syncthreads();

  // --- GEMM: each wave owns N-tiles wave, wave+16, ... (125 tiles) ---
  const int NTILES = V_ / 16;            // 2000
  for (int nt = wave; nt < NTILES; nt += 16) {
    const int n0 = nt * 16;
    // A frag: lanes 0-15 hold K={kk,kk+1} of row r; lanes 16-31 K={kk+2,kk+3}
    const float* arow = &ldsA[r][2 * half];
    // B frag (4x16, K-major): mirrored layout, column v = n0 + r
    const float* wrow = Wout + (size_t)(n0 + r) * H_ + 2 * half;

    // Prefetch this lane's row for the wave's next tile (global_prefetch_b8).
    if (n0 + 256 + r < V_) __builtin_prefetch(Wout + (size_t)(n0 + 256 + r) * H_, 0, 0);

    v8f acc = {0.f, 0.f, 0.f, 0.f, 0.f, 0.f, 0.f, 0.f};
    #pragma unroll 4
    for (int kk = 0; kk < H_; kk += 4) {
      v2f a  = *(const v2f*)(arow + kk);
      v2f bf = *(const v2f*)(wrow + kk);
      acc = __builtin_amdgcn_wmma_f32_16x16x4_f32(
          /*neg_a=*/false, a, /*neg_b=*/false, bf,
          /*c_mod=*/(short)0, acc, /*reuse_a=*/false, /*reuse_b=*/false);
    }

    // C layout: VGPR i -> row (i + 8*half), col = n0 + r. Store biased logits.
    const float bias = bout[n0 + r];
    #pragma unroll
    for (int i = 0; i < 8; ++i) {
      out[(size_t)(m0 + i + 8 * half) * V_ + (n0 + r)] = acc[i] + bias;
    }
  }

  __threadfence();                       // logits visible workgroup-wide
  __syncthreads();

  // --- Stats: wave w owns row w (L2-resident re-read, coalesced) ---
  {
    const size_t base = (size_t)(m0 + wave) * V_;
    float m = -INFINITY;
    for (int c = lane; c < V_; c += 32) m = fmaxf(m, out[base + c]);
    #pragma unroll
    for (int off = 16; off > 0; off >>= 1) m = fmaxf(m, __shfl_xor(m, off));
    float s = 0.f;
    for (int c = lane; c < V_; c += 32) s += __expf(out[base + c] - m);
    #pragma unroll
    for (int off = 16; off > 0; off >>= 1) s += __shfl_xor(s, off);
    if (lane == 0) rowlogz[wave] = m + __logf(s);
  }
  __syncthreads();

  // --- Normalize: log_probs = logit - logZ (wave per row, in place) ---
  {
    const float  lz   = rowlogz[wave];
    const size_t base = (size_t)(m0 + wave) * V_;
    for (int c = lane; c < V_; c += 32) out[base + c] -= lz;
  }
}

// ---------------------------------------------------------------------------
extern "C" void kernel_launch(void* const* d_in, const int* in_sizes, int n_in,
                              void* d_out, int out_size, void* d_ws, size_t ws_size,
                              hipStream_t stream) {
  (void)in_sizes; (void)n_in; (void)out_size; (void)ws_size;
  // setup_inputs order: 0=encoder_outputs (unused by reference), 1=encoder_hidden,
  // 2=target_tensor, 3=embedding, 4..7 layer0 W/b, 8..11 layer1 W/b, 12=W_out, 13=b_out
  const float* enc_hidden = (const float*)d_in[1];
  const int*   target     = (const int*)  d_in[2];
  const float* emb        = (const float*)d_in[3];
  const float* Wih0 = (const float*)d_in[4];
  const float* Whh0 = (const float*)d_in[5];
  const float* bih0 = (const float*)d_in[6];
  const float* bhh0 = (const float*)d_in[7];
  const float* Wih1 = (const float*)d_in[8];
  const float* Whh1 = (const float*)d_in[9];
  const float* bih1 = (const float*)d_in[10];
  const float* bhh1 = (const float*)d_in[11];
  const float* Wout = (const float*)d_in[12];
  const float* bout = (const float*)d_in[13];

  float* out  = (float*)d_out;                       // [B,T,V] ++ [L,B,H]
  float* tops = (float*)d_ws;                        // [T,B,H] scratch (4 MB)

  gru_seq_kernel<<<B_, H_, 0, stream>>>(enc_hidden, target, emb,
                                        Wih0, Whh0, bih0, bhh0,
                                        Wih1, Whh1, bih1, bhh1,
                                        tops, out + (size_t)B_ * T_ * V_);

  proj_logsoftmax_kernel<<<(T_ * B_) / 16, 512, 0, stream>>>(tops, Wout, bout, out);
}